// Embedding_17918603559083
// MI455X (gfx1250) — compile-verified
//
#include <hip/hip_runtime.h>

// GEMM: out[3200,300] = one_hot[3200,30000] @ vectors[30000,300], all fp32.
// HBM-bound: 384MB A stream @ 23.3 TB/s ~= 18us floor; vectors (36MB) stay in
// the 192MB L2. fp32 WMMA (V_WMMA_F32_16X16X4_F32) is bit-exact vs reference.
// A is streamed HBM->LDS by the Tensor Data Mover (TENSOR_LOAD_TO_LDS, one
// descriptor per 32x256 tile, TENSORcnt), double-buffered so the DMA of block
// k+1 overlaps the WMMAs of block k. TDM pad_enable skews LDS rows by 4 DWORDs
// (stride 260) so WMMA A-fragment column reads are LDS bank-conflict-free.

typedef float v2f __attribute__((ext_vector_type(2)));
typedef float v8f __attribute__((ext_vector_type(8)));
typedef uint32_t u32x4 __attribute__((ext_vector_type(4)));
typedef int      i32x8 __attribute__((ext_vector_type(8)));
typedef int      i32x4 __attribute__((ext_vector_type(4)));

#define M_TOT 3200
#define K_TOT 30000
#define N_TOT 300
#define MB     32          // M rows per workgroup (2 WMMA M-tiles per wave)
#define KC     256         // K block per TDM tile (pad_interval must be 2^n dwords)
#define LDS_STRIDE 260     // 256 + 4 pad dwords -> bank step 4/row, conflict-free
#define NWAVES 19          // ceil(300 / 16) N-tiles, one wave each
#define TPB (NWAVES * 32)

__device__ __forceinline__ void tensor_wait_zero() {
#if __has_builtin(__builtin_amdgcn_s_wait_tensorcnt)
    __builtin_amdgcn_s_wait_tensorcnt(0);
#else
    asm volatile("s_wait_tensorcnt 0x0" ::: "memory");
#endif
}

// Issue one TDM descriptor: load a [MB x KC] fp32 tile (row stride K_TOT) from
// global into LDS with +4-DWORD row padding and OOB-zero-fill past rem_k.
__device__ __forceinline__ void tdm_load_tile(const float* gtile,
                                              uint32_t lds_byte_off,
                                              uint32_t rem_k) {
    const uint64_t ga = (uint64_t)(uintptr_t)gtile;

    u32x4 g0;
    g0.x = 0x1u;                                        // count=1, user mode
    g0.y = lds_byte_off;                                // lds_addr (bytes)
    g0.z = (uint32_t)ga;                                // global_addr[31:0]
    g0.w = (uint32_t)((ga >> 32) & 0x1FFFFFFu)          // global_addr[56:32]
         | (2u << 30);                                  // type=2 ("image")

    i32x8 g1;
    g1[0] = (2 << 16)      // data_size = 4B
          | (1 << 20)      // pad_enable
          | (7 << 22)      // pad_interval: 256 DWORDs
          | (3 << 25);     // pad_amount: 4 DWORDs
    g1[1] = (int)((rem_k & 0xFFFFu) << 16);             // tensor_dim0[15:0]
    g1[2] = (int)((rem_k >> 16) & 0xFFFFu)              // tensor_dim0[31:16]
          | (MB << 16);                                 // tensor_dim1[15:0] = 32
    g1[3] = (KC << 16);                                 // tile_dim0 = 256
    g1[4] = MB;                                         // tile_dim1 = 32, tile_dim2 = 0
    g1[5] = K_TOT;                                      // tensor_dim0_stride[31:0]
    g1[6] = 0;                                          // stride hi / dim1_stride lo
    g1[7] = 0;

    const i32x4 gz = {0, 0, 0, 0};                      // groups 2/3: dims unused

    // Inline asm: portable across the 5-arg/6-arg builtin toolchain split.
    asm volatile("tensor_load_to_lds %0, %1, %2, %3"
                 :: "s"(g0), "s"(g1), "s"(gz), "s"(gz)
                 : "memory");
}

__global__ __launch_bounds__(TPB) void onehot_gemm_wmma(
    const float* __restrict__ A,    // [M_TOT, K_TOT] one_hot
    const float* __restrict__ Bm,   // [K_TOT, N_TOT] vectors
    float* __restrict__ C)          // [M_TOT, N_TOT]
{
    __shared__ float lds_a[2][MB * LDS_STRIDE];         // 2 x 33.3 KB

    const int tid   = threadIdx.x;
    const int wave  = tid >> 5;
    const int lane  = tid & 31;
    const int m0    = blockIdx.x * MB;
    const int n0    = wave * 16;
    const int ncol  = n0 + (lane & 15);
    const bool nok  = (ncol < N_TOT);
    const int nc    = nok ? ncol : (N_TOT - 1);         // clamp: loads in-bounds
    const int khalf = (lane < 16) ? 0 : 2;              // WMMA f32 half-wave K split

    v8f acc0 = {};  // C rows m0 .. m0+15
    v8f acc1 = {};  // C rows m0+16 .. m0+31

    // ---- prologue: TDM-stage block 0 into buffer 0 (wave 0 issues) ----
    if (wave == 0) {
        tdm_load_tile(A + (size_t)m0 * K_TOT,
                      (uint32_t)(uintptr_t)&lds_a[0][0], K_TOT);
    }

    int buf = 0;
    for (int kb = 0; kb < K_TOT; kb += KC, buf ^= 1) {
        // Wave 0's TDM op done -> barrier publishes the tile to all waves and
        // guarantees everyone finished reading the other buffer.
        if (wave == 0) tensor_wait_zero();
        __syncthreads();

        // ---- issue next tile into the other buffer (overlaps compute) ----
        const int kn = kb + KC;
        if (kn < K_TOT) {
            if (wave == 0) {
                tdm_load_tile(A + (size_t)m0 * K_TOT + kn,
                              (uint32_t)(uintptr_t)&lds_a[buf ^ 1][0],
                              (uint32_t)(K_TOT - kn));
            }
            // warm L2 one block further ahead (global_prefetch_b8)
            if (wave == 1 && kn + KC < K_TOT)
                __builtin_prefetch(A + (size_t)(m0 + lane) * K_TOT + kn + KC, 0, 0);
        }

        // ---- 64 WMMA K-steps over the staged tile ----
        const float* lbuf = &lds_a[buf][0];
        for (int kk = 0; kk < KC; kk += 4) {
            // B fragment 4x16 from L2-resident vectors. For the zero-filled K
            // tail, A-fragments are exactly 0, so clamped B rows contribute 0.
            int krow = kb + kk + khalf;
            krow = (krow < K_TOT - 1) ? krow : (K_TOT - 2);
            float b0 = Bm[(size_t)krow * N_TOT + nc];
            float b1 = Bm[(size_t)(krow + 1) * N_TOT + nc];
            v2f bfrag;
            bfrag.x = nok ? b0 : 0.0f;
            bfrag.y = nok ? b1 : 0.0f;

            // A fragment 16x4: lane m holds K = kk+khalf, +1 (one ds b64 read)
            v2f a0 = *(const v2f*)(&lbuf[(lane & 15) * LDS_STRIDE + kk + khalf]);
            acc0 = __builtin_amdgcn_wmma_f32_16x16x4_f32(
                false, a0, false, bfrag, (short)0, acc0, false, false);

            v2f a1 = *(const v2f*)(&lbuf[(16 + (lane & 15)) * LDS_STRIDE + kk + khalf]);
            acc1 = __builtin_amdgcn_wmma_f32_16x16x4_f32(
                false, a1, false, bfrag, (short)0, acc1, false, false);
        }
    }

    // ---- store: C VGPR r -> M = r (lanes 0-15) / r+8 (lanes 16-31) ----
    if (nok) {
        const int mbase = m0 + ((lane < 16) ? 0 : 8);
        #pragma unroll
        for (int r = 0; r < 8; ++r) {
            C[(size_t)(mbase + r) * N_TOT + ncol]      = acc0[r];
            C[(size_t)(mbase + 16 + r) * N_TOT + ncol] = acc1[r];
        }
    }
}

extern "C" void kernel_launch(void* const* d_in, const int* in_sizes, int n_in,
                              void* d_out, int out_size, void* d_ws, size_t ws_size,
                              hipStream_t stream) {
    const float* one_hot = (const float*)d_in[0];   // [64*50, 30000]
    const float* vectors = (const float*)d_in[1];   // [30000, 300]
    float* out = (float*)d_out;                     // [64*50, 300]

    dim3 grid(M_TOT / MB);   // 100 workgroups
    dim3 block(TPB);         // 19 waves of 32
    onehot_gemm_wmma<<<grid, block, 0, stream>>>(one_hot, vectors, out);
}